// CorrelationLayer2D_13580686590317
// MI455X (gfx1250) — compile-verified
//
#include <hip/hip_runtime.h>
#include <stdint.h>

typedef __attribute__((ext_vector_type(2))) float v2f;
typedef __attribute__((ext_vector_type(8))) float v8f;
typedef __attribute__((ext_vector_type(4))) unsigned int u32x4;
typedef __attribute__((ext_vector_type(8))) int i32x8;
typedef __attribute__((ext_vector_type(4))) int i32x4;

#define MD   20
#define ND   41            // 2*MD+1 displacements per axis
#define C_   256
#define H_   64
#define W_   128
#define KCH  16            // channels per LDS chunk (double-buffered)
#define ASTR 136           // A LDS row stride (128 data + 8 pad DW) -> bank shift 16
#define BSTR 176           // B LDS row stride (128 data + 48 pad DW) -> bank shift 32
#define GSTR 44            // Gl dy-stride (>= ND, padded)

#define ABUF (KCH * ASTR * 4)            //  8704 B per A buffer
#define BBUF (KCH * BSTR * 4)            // 11264 B per B buffer
#define A_OFF 0
#define B_OFF (2 * ABUF)                 // 17408 B
#define G_OFF (B_OFF + 2 * BBUF)         // 39936 B
#define DYN_LDS (G_OFF + W_ * GSTR * 4)  // 62464 B dynamic LDS

#define HAVE_TDM __has_builtin(__builtin_amdgcn_tensor_load_to_lds)

#if HAVE_TDM
// Issue one 2-D TDM tile load: `rows` rows of `tile_w` fp32, global row stride
// `row_stride_elems`, written to LDS at `lds_byte_off` with `pad_dw` pad DWORDs
// appended after every 128 data DWORDs (pad_interval code 6 == 128 DW).
__device__ __forceinline__ void tdm_load_2d(uint32_t lds_byte_off, const float* gptr,
                                            int tile_w, int rows,
                                            int row_stride_elems, int pad_dw)
{
    uint64_t ga = (uint64_t)(uintptr_t)gptr;
    u32x4 g0;
    g0[0] = 1u;                                  // D# count = 1 (valid), user mode
    g0[1] = lds_byte_off;                        // lds_addr [63:32]
    g0[2] = (uint32_t)ga;                        // global_addr low  [95:64]
    g0[3] = (uint32_t)(ga >> 32) | (2u << 30);   // global_addr high | type=2 (image)

    i32x8 g1;
    int f = (2 << 16);                           // data_size = 2 -> 4 bytes
    if (pad_dw > 0)
        f |= (1 << 20)                           // pad_enable
           | (6 << 22)                           // pad_interval: 128 DWORDs
           | ((pad_dw - 1) << 25);               // pad_amount: pad_dw DWORDs
    g1[0] = f;                                   // workgroup_mask=0, no barrier/iter
    g1[1] = (tile_w & 0xffff) << 16;             // tensor_dim0[15:0]   (bits 79:48)
    g1[2] = ((tile_w >> 16) & 0xffff)            // tensor_dim0[31:16]
          | ((rows & 0xffff) << 16);             // tensor_dim1[15:0]
    g1[3] = ((rows >> 16) & 0xffff)              // tensor_dim1[31:16]
          | ((tile_w & 0xffff) << 16);           // tile_dim0 (bits 127:112)
    g1[4] = (rows & 0xffff);                     // tile_dim1 | tile_dim2=0
    g1[5] = row_stride_elems;                    // tensor_dim0_stride[31:0]
    g1[6] = 0;                                   // stride hi | tensor_dim1_stride lo
    g1[7] = 0;
    i32x4 gz4 = {0, 0, 0, 0};                    // groups 2/3: unused (2-D tensor)
    i32x8 gz8 = {0, 0, 0, 0, 0, 0, 0, 0};        // trailing group (clang-23 form)
    __builtin_amdgcn_tensor_load_to_lds(g0, g1, gz4, gz4, gz8, 0);
}
#endif

// out[b, dx*41+dy, h, w] = sum_c x[b,c,h,w] * x[b,c,h+dx-20,w+dy-20]
// Banded GEMM per (b,h,dx): G[m,j] = sum_c A[c,w0+m]*Brow[c,j],
// out[dy, w0+m] = G[m, m+dy], on v_wmma_f32_16x16x4_f32, with
// double-buffered TDM (tensor_load_to_lds) staging overlapped with compute.
__global__ __launch_bounds__(256)
void corr2d_wmma_tdm(const float* __restrict__ X, float* __restrict__ out)
{
    extern __shared__ float smem[];
    float* Gl = smem + G_OFF / 4;      // [W_][GSTR] output staging

    const int dxi = blockIdx.x;        // 0..40
    const int h   = blockIdx.y;        // 0..63
    const int b   = blockIdx.z;        // 0..B-1

    const int tid  = threadIdx.x;
    const int wave = tid >> 5;         // 0..7 -> w-tile
    const int lane = tid & 31;
    const int mrow = lane & 15;        // M (A) / N (B,D) index
    const int nsub = lane >> 4;        // K-pair select / D row half
    const int koff = nsub << 1;        // A/B K offset: 0 or 2
    const int w0   = wave << 4;
    const int acol = w0 + mrow;

    const int  hsrc = h + dxi - MD;
    const bool hok  = (hsrc >= 0) && (hsrc < H_);

    const float* Arow = X + (((size_t)b * C_) * H_ + (size_t)h) * W_;
    const float* Brow = X + (((size_t)b * C_) * H_ + (size_t)(hok ? hsrc : 0)) * W_;

    // Zero both B staging buffers once: halo columns (j<20, j>=148) stay zero
    // for all chunks; if the source row is out of range B stays entirely zero.
    for (int i = tid; i < 2 * KCH * BSTR; i += 256) smem[B_OFF / 4 + i] = 0.0f;

#if HAVE_TDM
    // Prologue: start chunk 0 DMA into buffer 0 (disjoint from halo zeros).
    if (wave == 0) {
        tdm_load_2d(A_OFF, Arow, W_, KCH, H_ * W_, ASTR - W_);
        if (hok)
            tdm_load_2d(B_OFF + MD * 4, Brow, W_, KCH, H_ * W_, BSTR - W_);
    }
#endif

    v8f acc[4] = {};   // 4 N-tiles: D columns j0 = w0 + 16n

    for (int k0 = 0; k0 < C_; k0 += KCH) {
        const int p = (k0 / KCH) & 1;                  // buffer parity of chunk
        const float* As = smem + (A_OFF + p * ABUF) / 4;
        const float* Bs = smem + (B_OFF + p * BBUF) / 4;

        __syncthreads();   // parity p^1 reads (chunk k0-KCH) complete
#if HAVE_TDM
        if (wave == 0) {
            if (k0 + KCH < C_) {
                const int np = p ^ 1;
                // Kick off next chunk's DMA, then wait only for THIS chunk:
                // tensor ops complete in order, so tensorcnt <= n_just_issued
                // means everything issued earlier has landed in LDS.
                tdm_load_2d(A_OFF + np * ABUF,
                            Arow + (size_t)(k0 + KCH) * (H_ * W_),
                            W_, KCH, H_ * W_, ASTR - W_);
                if (hok) {
                    tdm_load_2d(B_OFF + np * BBUF + MD * 4,
                                Brow + (size_t)(k0 + KCH) * (H_ * W_),
                                W_, KCH, H_ * W_, BSTR - W_);
                    __builtin_amdgcn_s_wait_tensorcnt(2);
                } else {
                    __builtin_amdgcn_s_wait_tensorcnt(1);
                }
            } else {
                __builtin_amdgcn_s_wait_tensorcnt(0);
            }
        }
#else
        {   // synchronous fallback staging into parity p
            float* Asw = smem + (A_OFF + p * ABUF) / 4;
            float* Bsw = smem + (B_OFF + p * BBUF) / 4;
            for (int i = tid; i < KCH * W_; i += 256) {
                int kk = i >> 7, w = i & (W_ - 1);
                Asw[kk * ASTR + w] = Arow[(size_t)(k0 + kk) * (H_ * W_) + w];
            }
            if (hok) {
                for (int i = tid; i < KCH * W_; i += 256) {
                    int kk = i >> 7, w = i & (W_ - 1);
                    Bsw[kk * BSTR + MD + w] = Brow[(size_t)(k0 + kk) * (H_ * W_) + w];
                }
            }
        }
#endif
        __syncthreads();   // chunk k0 visible in parity p

        #pragma unroll
        for (int kk = 0; kk < KCH; kk += 4) {
            // A frag: lanes 0-15 = (M=mrow, K=kk,kk+1); lanes 16-31 = (K+2,K+3)
            v2f a;
            a.x = As[(kk + koff) * ASTR + acol];
            a.y = As[(kk + koff + 1) * ASTR + acol];
            #pragma unroll
            for (int n = 0; n < 4; ++n) {
                int bcol = w0 + (n << 4) + mrow;
                v2f bf;
                bf.x = Bs[(kk + koff) * BSTR + bcol];
                bf.y = Bs[(kk + koff + 1) * BSTR + bcol];
                acc[n] = __builtin_amdgcn_wmma_f32_16x16x4_f32(
                    false, a, false, bf, (short)0, acc[n], false, false);
            }
        }
    }

    __syncthreads();
    // Scatter diagonal band G[m, m+dy] -> Gl[w][dy]
    #pragma unroll
    for (int n = 0; n < 4; ++n) {
        #pragma unroll
        for (int r = 0; r < 8; ++r) {
            int w  = w0 + r + (nsub << 3);               // D row
            int dy = (n << 4) + mrow - r - (nsub << 3);  // j - w
            if ((unsigned)dy < (unsigned)ND)
                Gl[w * GSTR + dy] = acc[n][r];
        }
    }
    __syncthreads();

    // Coalesced store: 41 contiguous 512B rows of out[b, dx*41+dy, h, :]
    const size_t obase =
        (((size_t)b * (ND * ND) + (size_t)dxi * ND) * H_ + (size_t)h) * W_;
    for (int i = tid; i < ND * W_; i += 256) {
        int dyi = i >> 7;
        int w   = i & (W_ - 1);
        out[obase + (size_t)dyi * (H_ * W_) + w] = Gl[w * GSTR + dyi];
    }
}

extern "C" void kernel_launch(void* const* d_in, const int* in_sizes, int n_in,
                              void* d_out, int out_size, void* d_ws, size_t ws_size,
                              hipStream_t stream) {
    const float* x   = (const float*)d_in[0];
    float*       out = (float*)d_out;
    const int B = in_sizes[0] / (C_ * H_ * W_);   // = 4
    dim3 grid(ND, H_, B);
    dim3 block(256);
    hipLaunchKernelGGL(corr2d_wmma_tdm, grid, block, DYN_LDS, stream, x, out);
}